// GCNLayer_20031727468913
// MI455X (gfx1250) — compile-verified
//
#include <hip/hip_runtime.h>

#define EMB 64

typedef __attribute__((ext_vector_type(2))) float v2f;
typedef __attribute__((ext_vector_type(8))) float v8f;

__device__ __forceinline__ void atomAddF(float* p, float v) {
    // hardware global_atomic_add_f32 (L2-resident targets; no CAS loop)
    unsafeAtomicAdd(p, v);
}

// ---------------- zero workspace (degrees) ----------------
__global__ void zero_f32_kernel(float* __restrict__ p, int n) {
    int i = blockIdx.x * blockDim.x + threadIdx.x;
    if (i < n) p[i] = 0.0f;
}

// ---------------- degree histogram ----------------
__global__ void degree_kernel(const int* __restrict__ uidx, const int* __restrict__ iidx,
                              float* __restrict__ deg_u, float* __restrict__ deg_i, int E) {
    int e = blockIdx.x * blockDim.x + threadIdx.x;
    if (e < E) {
        atomAddF(&deg_u[uidx[e]], 1.0f);
        atomAddF(&deg_i[iidx[e]], 1.0f);
    }
}

// ---------------- edge scatter: one wave (32 lanes) per edge ----------------
// out_u/out_i already hold the residual (u_emb / i_emb copies).
__global__ void scatter_kernel(const float* __restrict__ u_emb, const float* __restrict__ i_emb,
                               const int* __restrict__ uidx, const int* __restrict__ iidx,
                               const float* __restrict__ wts,
                               const float* __restrict__ deg_u, const float* __restrict__ deg_i,
                               float* __restrict__ out_u, float* __restrict__ out_i, int E) {
    int wave = (blockIdx.x * blockDim.x + threadIdx.x) >> 5;
    if (wave >= E) return;
    int lane = threadIdx.x & 31;

    int u  = uidx[wave];
    int it = iidx[wave];
    float du = fmaxf(deg_u[u], 1.0f);
    float di = fmaxf(deg_i[it], 1.0f);
    float w  = wts[wave] / (sqrtf(du) * sqrtf(di));

    int k = lane * 2;  // each lane: 2 consecutive dims of the 64-dim row
    v2f iv = *(const v2f*)(i_emb + (size_t)it * EMB + k);
    v2f uv = *(const v2f*)(u_emb + (size_t)u  * EMB + k);

    atomAddF(out_u + (size_t)u  * EMB + k,     iv.x * w);
    atomAddF(out_u + (size_t)u  * EMB + k + 1, iv.y * w);
    atomAddF(out_i + (size_t)it * EMB + k,     uv.x * w);
    atomAddF(out_i + (size_t)it * EMB + k + 1, uv.y * w);
}

// ---------------- in-place Y = relu(X @ W^T), X:[nrows,64], W:[64,64] ----------------
// One wave per 16-row block. Uses V_WMMA_F32_16X16X4_F32 (full f32 precision).
// A-tile layout (16x4 f32): lane L<16 -> row L, {K+0,K+1}; lane L+16 -> row L, {K+2,K+3}
//   => float2 load at X[(r0+lm)*64 + kbase + 2*hi]
// B-tile (4x16 f32), B[k][n] = W[n0+n][kbase+k]: symmetric layout
//   => float2 load at W[(n0+lm)*64 + kbase + 2*hi]
__global__ void gemm_relu_inplace(float* __restrict__ X, const float* __restrict__ W, int nrows) {
    int wave = (blockIdx.x * blockDim.x + threadIdx.x) >> 5;
    int r0 = wave * 16;
    if (r0 >= nrows) return;   // wave-uniform: EXEC stays all-1s for WMMA

    int lane = threadIdx.x & 31;
    int lm = lane & 15;
    int hi = lane >> 4;

    // Load full A block (16 rows x 64 K) once: 16 K-steps of float2 per lane.
    v2f a[16];
    const float* arow = X + (size_t)(r0 + lm) * EMB + 2 * hi;
#pragma unroll
    for (int ks = 0; ks < 16; ++ks)
        a[ks] = *(const v2f*)(arow + ks * 4);

    v8f acc[4];
#pragma unroll
    for (int t = 0; t < 4; ++t) {
        v8f c = {};
        const float* wbase = W + (size_t)(t * 16 + lm) * EMB + 2 * hi;
#pragma unroll
        for (int ks = 0; ks < 16; ++ks) {
            v2f b = *(const v2f*)(wbase + ks * 4);
            c = __builtin_amdgcn_wmma_f32_16x16x4_f32(
                    /*neg_a=*/false, a[ks], /*neg_b=*/false, b,
                    /*c_mod=*/(short)0, c, /*reuse_a=*/false, /*reuse_b=*/false);
        }
        acc[t] = c;
    }

    // ReLU + store. D layout: VGPR v, lane half -> row r0 + v + 8*hi, col t*16 + lm.
    // All reads of this wave's rows happened above; rows are wave-exclusive -> in-place safe.
#pragma unroll
    for (int t = 0; t < 4; ++t) {
#pragma unroll
        for (int v = 0; v < 8; ++v) {
            float y = fmaxf(acc[t][v], 0.0f);
            X[(size_t)(r0 + v + 8 * hi) * EMB + t * 16 + lm] = y;
        }
    }
}

extern "C" void kernel_launch(void* const* d_in, const int* in_sizes, int n_in,
                              void* d_out, int out_size, void* d_ws, size_t ws_size,
                              hipStream_t stream) {
    const float* u_emb = (const float*)d_in[0];
    const float* i_emb = (const float*)d_in[1];
    const int*   edge  = (const int*)d_in[2];
    const float* wts   = (const float*)d_in[3];
    const float* W_u   = (const float*)d_in[4];
    const float* W_i   = (const float*)d_in[5];

    const int U = in_sizes[0] / EMB;   // 500000
    const int I = in_sizes[1] / EMB;   // 200000
    const int E = in_sizes[3];         // 1000000

    const int* uidx = edge;        // edge_index[0]
    const int* iidx = edge + E;    // edge_index[1]

    float* out_u = (float*)d_out;
    float* out_i = out_u + (size_t)U * EMB;

    float* deg_u = (float*)d_ws;   // U floats
    float* deg_i = deg_u + U;      // I floats

    // 1) zero degree scratch (must be re-done every call)
    int nz = U + I;
    zero_f32_kernel<<<(nz + 255) / 256, 256, 0, stream>>>(deg_u, nz);

    // 2) degree histograms
    degree_kernel<<<(E + 255) / 256, 256, 0, stream>>>(uidx, iidx, deg_u, deg_i, E);

    // 3) residual init: d_out <- [u_emb ; i_emb]
    hipMemcpyAsync(out_u, u_emb, (size_t)U * EMB * sizeof(float),
                   hipMemcpyDeviceToDevice, stream);
    hipMemcpyAsync(out_i, i_emb, (size_t)I * EMB * sizeof(float),
                   hipMemcpyDeviceToDevice, stream);

    // 4) edge scatter (one wave per edge; 8 waves / 256-thread block)
    scatter_kernel<<<(E + 7) / 8, 256, 0, stream>>>(u_emb, i_emb, uidx, iidx, wts,
                                                    deg_u, deg_i, out_u, out_i, E);

    // 5) in-place GEMM + ReLU with f32 WMMA (one wave per 16 rows; 8 waves/block)
    int wu = (U + 15) / 16;
    gemm_relu_inplace<<<(wu + 7) / 8, 256, 0, stream>>>(out_u, W_u, U);
    int wi = (I + 15) / 16;
    gemm_relu_inplace<<<(wi + 7) / 8, 256, 0, stream>>>(out_i, W_i, I);
}